// SimpleGCN_14714557956354
// MI455X (gfx1250) — compile-verified
//
#include <hip/hip_runtime.h>

typedef __attribute__((ext_vector_type(2))) float v2f;
typedef __attribute__((ext_vector_type(8))) float v8f;

#define NNODES 100000
#define NEDGES 1600000
#define NGRAPH 128
#define HID    128
#define NCLS   16

// ---------------- zero workspace ----------------
__global__ void k_zero(float* __restrict__ p, long long n) {
  long long i = (long long)blockIdx.x * blockDim.x + threadIdx.x;
  if (i < n) p[i] = 0.0f;
}

// ---------------- degree counts ----------------
__global__ void k_deg(const int* __restrict__ src, const int* __restrict__ dst,
                      float* __restrict__ in_deg, float* __restrict__ out_deg, int E) {
  int e = blockIdx.x * blockDim.x + threadIdx.x;
  if (e < E) {
    atomicAdd(&in_deg[dst[e]], 1.0f);
    atomicAdd(&out_deg[src[e]], 1.0f);
  }
}

// ---------------- per-node: inv sqrt norms, layer-1 input, graph counts ----------------
__global__ void k_node1(const float* __restrict__ in_deg, const float* __restrict__ out_deg,
                        const int* __restrict__ gid,
                        float* __restrict__ inv_in, float* __restrict__ inv_out,
                        float* __restrict__ xs0, float* __restrict__ cnt, int N) {
  int n = blockIdx.x * blockDim.x + threadIdx.x;
  if (n < N) {
    float di = in_deg[n], dO = out_deg[n];
    float ii = di > 0.0f ? rsqrtf(di) : 0.0f;
    float io = dO > 0.0f ? rsqrtf(dO) : 0.0f;
    inv_in[n] = ii;
    inv_out[n] = io;
    xs0[n] = di * io;          // h0 = in_deg; xs0 = h0 * inv_out
    atomicAdd(&cnt[gid[n]], 1.0f);
  }
}

// ---------------- layer-1 edge aggregation (scalar feature) ----------------
__global__ void k_edge1(const int* __restrict__ src, const int* __restrict__ dst,
                        const float* __restrict__ xs0, float* __restrict__ agg1, int E) {
  int e = blockIdx.x * blockDim.x + threadIdx.x;
  if (e < E) atomicAdd(&agg1[dst[e]], xs0[src[e]]);
}

// ---------------- layer-1 dense: h1 = relu(a1*W1 + b1); store xs1 = h1*inv_out ----------------
__global__ void k_layer1(const float* __restrict__ agg1, const float* __restrict__ inv_in,
                         const float* __restrict__ inv_out,
                         const float* __restrict__ W1, const float* __restrict__ b1,
                         float* __restrict__ xs1, int N) {
  long long idx = (long long)blockIdx.x * blockDim.x + threadIdx.x;
  if (idx < (long long)N * HID) {
    int n = (int)(idx >> 7);
    int j = (int)(idx & (HID - 1));
    float a = agg1[n] * inv_in[n];
    float h = a * W1[j] + b1[j];
    h = h > 0.0f ? h : 0.0f;
    xs1[idx] = h * inv_out[n];
  }
}

// ---------------- layer-2 edge aggregation: one wave per edge, float4 per lane ----------------
__global__ void k_edge2(const int* __restrict__ src, const int* __restrict__ dst,
                        const float* __restrict__ xs1, float* __restrict__ agg2, int E) {
  long long tid = (long long)blockIdx.x * blockDim.x + threadIdx.x;
  int e = (int)(tid >> 5);
  if (e >= E) return;
  int jb = ((int)tid & 31) * 4;           // 32 lanes x 4 floats = 128 features
  int s = src[e], d = dst[e];
  const float4 v = *(const float4*)(xs1 + (long long)s * HID + jb);
  float* o = agg2 + (long long)d * HID + jb;
  atomicAdd(o + 0, v.x);
  atomicAdd(o + 1, v.y);
  atomicAdd(o + 2, v.z);
  atomicAdd(o + 3, v.w);
}

// ---------------- pack W2 into B-fragment order: P[k/2][c] = (W2[k][c], W2[k+1][c]) ----------------
__global__ void k_packW2(const float* __restrict__ W2, float2* __restrict__ P) {
  int idx = blockIdx.x * blockDim.x + threadIdx.x;     // 64*128 float2 elements
  if (idx < (HID / 2) * HID) {
    int kp = idx >> 7;                                  // K pair index (0..63)
    int c = idx & (HID - 1);
    float2 v;
    v.x = W2[(2 * kp) * HID + c];
    v.y = W2[(2 * kp + 1) * HID + c];
    P[idx] = v;
  }
}

// ---------------- layer-2 GEMM (f32 WMMA 16x16x4) fused with ReLU + graph pooling ----------------
// Block = 256 threads = 8 waves; block handles 16 nodes; wave w handles cols [16w, 16w+16).
__global__ __launch_bounds__(256) void
k_gemm2_pool(const float* __restrict__ agg2, const float* __restrict__ inv_in,
             const float2* __restrict__ Bpk, const float* __restrict__ b2,
             const int* __restrict__ gid, float* __restrict__ gsum) {
  __shared__ float As[16 * 130];          // 16 rows x 128 K, stride 130 (bank-conflict pad)

  const int nodeBase = blockIdx.x * 16;

  // Stage A = agg2 * inv_in for 16 nodes into LDS (float2 loads/stores, 8B aligned).
  for (int i = threadIdx.x; i < 16 * (HID / 2); i += 256) {
    int row = i >> 6;                     // 64 float2 per row
    int kp = (i & 63) * 2;
    int node = nodeBase + row;
    float sc = inv_in[node];
    float2 v = *(const float2*)(agg2 + (long long)node * HID + kp);
    *(float2*)(&As[row * 130 + kp]) = make_float2(v.x * sc, v.y * sc);
  }
  __syncthreads();

  const int lane = threadIdx.x & 31;
  const int wave = threadIdx.x >> 5;
  const int colBase = wave * 16;
  const int col = lane & 15;
  const int khalf = lane >> 4;            // lanes 0-15 -> K {0,1}; lanes 16-31 -> K {2,3}
  const int rowA = lane & 15;
  const int colIdx = colBase + col;

  v8f c = {};
  #pragma unroll
  for (int k0 = 0; k0 < HID; k0 += 4) {
    const int ka = k0 + 2 * khalf;        // even
    v2f a, b;
    a.x = As[rowA * 130 + ka];
    a.y = As[rowA * 130 + ka + 1];
    float2 bb = Bpk[(ka >> 1) * HID + colIdx];   // single global_load_b64
    b.x = bb.x;
    b.y = bb.y;
    c = __builtin_amdgcn_wmma_f32_16x16x4_f32(false, a, false, b, (short)0, c,
                                              false, false);
  }

  // Epilogue: bias + ReLU; run-length merge rows by graph id (sorted) before atomics.
  const float bias = b2[colIdx];
  const int rowOff = khalf * 8;           // C/D: lanes 0-15 rows 0-7, lanes 16-31 rows 8-15
  int gPrev = gid[nodeBase + rowOff];
  float acc = 0.0f;
  #pragma unroll
  for (int r = 0; r < 8; ++r) {
    int node = nodeBase + rowOff + r;
    float v = c[r] + bias;
    v = v > 0.0f ? v : 0.0f;
    int g = gid[node];
    if (g != gPrev) {
      atomicAdd(&gsum[(long long)gPrev * HID + colIdx], acc);
      acc = 0.0f;
      gPrev = g;
    }
    acc += v;
  }
  atomicAdd(&gsum[(long long)gPrev * HID + colIdx], acc);
}

// ---------------- classifier: out = (gsum/max(cnt,1)) @ Wl + bl ----------------
__global__ void k_out(const float* __restrict__ gsum, const float* __restrict__ cnt,
                      const float* __restrict__ Wl, const float* __restrict__ bl,
                      float* __restrict__ out) {
  int idx = blockIdx.x * blockDim.x + threadIdx.x;
  if (idx >= NGRAPH * NCLS) return;
  int g = idx >> 4;
  int cc = idx & (NCLS - 1);
  float ct = cnt[g];
  ct = ct > 1.0f ? ct : 1.0f;
  float s = 0.0f;
  #pragma unroll 8
  for (int k = 0; k < HID; ++k) s += gsum[g * HID + k] * Wl[k * NCLS + cc];
  out[idx] = s / ct + bl[cc];
}

extern "C" void kernel_launch(void* const* d_in, const int* in_sizes, int n_in,
                              void* d_out, int out_size, void* d_ws, size_t ws_size,
                              hipStream_t stream) {
  const int* src = (const int*)d_in[0];
  const int* dst = (const int*)d_in[1];
  const int* gid = (const int*)d_in[2];
  const float* W1 = (const float*)d_in[3];
  const float* b1 = (const float*)d_in[4];
  const float* W2 = (const float*)d_in[5];
  const float* b2 = (const float*)d_in[6];
  const float* Wl = (const float*)d_in[7];
  const float* bl = (const float*)d_in[8];
  float* out = (float*)d_out;

  const long long N = NNODES, E = NEDGES;

  // Workspace layout (floats)
  float* ws = (float*)d_ws;
  float* in_deg  = ws;                 // N
  float* out_deg = in_deg + N;         // N
  float* inv_in  = out_deg + N;        // N
  float* inv_out = inv_in + N;         // N
  float* agg1    = inv_out + N;        // N
  float* xs0     = agg1 + N;           // N
  float* xs1     = xs0 + N;            // N*HID
  float* agg2    = xs1 + N * HID;      // N*HID
  float* gsum    = agg2 + N * HID;     // NGRAPH*HID
  float* cnt     = gsum + NGRAPH * HID;// NGRAPH
  float* w2pk    = cnt + NGRAPH;       // HID*HID (packed float2 pairs)
  long long zeroTotal = 6 * N + 2 * N * HID + (long long)NGRAPH * HID + NGRAPH;

  k_zero<<<(int)((zeroTotal + 255) / 256), 256, 0, stream>>>(ws, zeroTotal);
  k_packW2<<<((HID / 2) * HID + 255) / 256, 256, 0, stream>>>(W2, (float2*)w2pk);
  k_deg<<<(int)((E + 255) / 256), 256, 0, stream>>>(src, dst, in_deg, out_deg, (int)E);
  k_node1<<<(int)((N + 255) / 256), 256, 0, stream>>>(in_deg, out_deg, gid,
                                                      inv_in, inv_out, xs0, cnt, (int)N);
  k_edge1<<<(int)((E + 255) / 256), 256, 0, stream>>>(src, dst, xs0, agg1, (int)E);
  k_layer1<<<(int)((N * HID + 255) / 256), 256, 0, stream>>>(agg1, inv_in, inv_out,
                                                             W1, b1, xs1, (int)N);
  k_edge2<<<(int)((E * 32 + 255) / 256), 256, 0, stream>>>(src, dst, xs1, agg2, (int)E);
  k_gemm2_pool<<<(int)(N / 16), 256, 0, stream>>>(agg2, inv_in, (const float2*)w2pk,
                                                  b2, gid, gsum);
  k_out<<<(NGRAPH * NCLS + 255) / 256, 256, 0, stream>>>(gsum, cnt, Wl, bl, out);
}